// Attention4_70841190580883
// MI455X (gfx1250) — compile-verified
//
#include <hip/hip_runtime.h>
#include <hip/hip_bf16.h>

#define B_ 16
#define L_ 2048   // 2^11
#define H_ 1024
#define Z_ 512
#define NEGINF (-1e9f)

typedef __attribute__((ext_vector_type(2))) float v2f;
typedef __attribute__((ext_vector_type(8))) float v8f;

// ---------------------------------------------------------------------------
// Small dense GEMM: C[16 x N] = A[16 x K] * B[K x N], one wave per 16-col tile.
// Uses V_WMMA_F32_16X16X4_F32.  TRANSB=1 -> B element (k,n) read at B[n*ldb+k].
// A layout (32-bit A 16x4): lanes 0-15 hold K=k0,k0+1; lanes 16-31 hold k0+2,k0+3.
// B layout (4x16, row striped across lanes): VGPR0 = rows k0 (lanes 0-15) /
// k0+2 (lanes 16-31), VGPR1 = k0+1 / k0+3.
// C/D layout: VGPR r -> M=r (lanes 0-15), M=r+8 (lanes 16-31), N = lane&15.
// ---------------------------------------------------------------------------
template <int TRANSB>
__global__ void wmma_gemm16(const float* __restrict__ A, int lda,
                            const float* __restrict__ Bm, int ldb,
                            float* __restrict__ C, int ldc, int K) {
  const int lane = threadIdx.x & 31;
  const int half = lane >> 4;
  const int l15  = lane & 15;
  const int n    = blockIdx.x * 16 + l15;
  v8f acc = {};
  for (int k0 = 0; k0 < K; k0 += 4) {
    const int ka = k0 + 2 * half;
    v2f a, b;
    a.x = A[l15 * lda + ka];
    a.y = A[l15 * lda + ka + 1];
    if (TRANSB) {
      b.x = Bm[n * ldb + ka];
      b.y = Bm[n * ldb + ka + 1];
    } else {
      b.x = Bm[ka * ldb + n];
      b.y = Bm[(ka + 1) * ldb + n];
    }
    acc = __builtin_amdgcn_wmma_f32_16x16x4_f32(false, a, false, b,
                                                (short)0, acc, false, false);
  }
#pragma unroll
  for (int r = 0; r < 8; ++r) C[(r + 8 * half) * ldc + n] = acc[r];
}

// ---------------------------------------------------------------------------
// Stage Z1: partial sums of z over L-chunks.  grid (Z/256, 16, B), block 256.
// partial[(b*16+chunk)*Z + z] = sum over 128 rows of z[b, :, z]
// ---------------------------------------------------------------------------
__global__ void zsum_partial(const float* __restrict__ z, float* __restrict__ partial) {
  const int zz  = blockIdx.x * 256 + threadIdx.x;
  const int lch = blockIdx.y;
  const int b   = blockIdx.z;
  const float* base = z + ((size_t)b * L_ + (size_t)lch * (L_ / 16)) * Z_ + zz;
  float acc = 0.0f;
#pragma unroll 4
  for (int j = 0; j < L_ / 16; ++j) acc += base[(size_t)j * Z_];
  partial[((size_t)(b * 16 + lch)) * Z_ + zz] = acc;
}

// Stage Z2: reduce 16 partials -> zsum[B,Z].  grid (Z/256, B), block 256.
__global__ void zsum_reduce(const float* __restrict__ partial, float* __restrict__ zsum) {
  const int zz = blockIdx.x * 256 + threadIdx.x;
  const int b  = blockIdx.y;
  float acc = 0.0f;
#pragma unroll
  for (int c = 0; c < 16; ++c) acc += partial[((size_t)(b * 16 + c)) * Z_ + zz];
  zsum[(size_t)b * Z_ + zz] = acc;
}

// ---------------------------------------------------------------------------
// align[b,i] = (enc[b,i] . kq[b]) / sqrt(H).  One wave32 per row, float4 loads.
// Dominant read: 134 MB of enc.
// ---------------------------------------------------------------------------
__global__ void align_kernel(const float* __restrict__ enc,
                             const float* __restrict__ kq,
                             float* __restrict__ algn) {
  const int row  = blockIdx.x * 8 + (threadIdx.x >> 5);   // 8 waves per block
  const int lane = threadIdx.x & 31;
  const int b    = row >> 11;                             // L = 2048
  const float* erow = enc + (size_t)row * H_;
  const float* qrow = kq + (size_t)b * H_;
  float acc = 0.0f;
#pragma unroll
  for (int h0 = 0; h0 < H_; h0 += 128) {
    if (h0 + 128 < H_)
      __builtin_prefetch(erow + h0 + 128 + lane * 4, 0, 1);  // global_prefetch_b8
    const int h = h0 + lane * 4;
    const float4 e = *(const float4*)(erow + h);
    const float4 q = *(const float4*)(qrow + h);
    acc = fmaf(e.x, q.x, fmaf(e.y, q.y, fmaf(e.z, q.z, fmaf(e.w, q.w, acc))));
  }
#pragma unroll
  for (int m = 16; m >= 1; m >>= 1) acc += __shfl_xor(acc, m, 32);
  if (lane == 0) algn[row] = acc * 0.03125f;   // 1/sqrt(1024) = 1/32
}

// ---------------------------------------------------------------------------
// Masked softmax over the query axis i, per batch.  One block per b.
// w[b,i] = softmax_i( mask[b,i] ? align[b,i] : -1e9 )
// ---------------------------------------------------------------------------
__global__ void softmax_kernel(const float* __restrict__ algn,
                               const int* __restrict__ mask,
                               float* __restrict__ w) {
  __shared__ float red[256];
  const int b = blockIdx.x, tid = threadIdx.x;
  float vals[L_ / 256];
#pragma unroll
  for (int r = 0; r < L_ / 256; ++r) {
    const int i = tid + r * 256;
    const float a = algn[(size_t)b * L_ + i];
    vals[r] = (mask[(size_t)b * L_ + i] == 0) ? NEGINF : a;
  }
  float mx = vals[0];
#pragma unroll
  for (int r = 1; r < L_ / 256; ++r) mx = fmaxf(mx, vals[r]);
  red[tid] = mx;
  __syncthreads();
  for (int s = 128; s > 0; s >>= 1) {
    if (tid < s) red[tid] = fmaxf(red[tid], red[tid + s]);
    __syncthreads();
  }
  mx = red[0];
  __syncthreads();
  float sum = 0.0f;
#pragma unroll
  for (int r = 0; r < L_ / 256; ++r) {
    vals[r] = expf(vals[r] - mx);
    sum += vals[r];
  }
  red[tid] = sum;
  __syncthreads();
  for (int s = 128; s > 0; s >>= 1) {
    if (tid < s) red[tid] += red[tid + s];
    __syncthreads();
  }
  const float inv = 1.0f / red[0];
#pragma unroll
  for (int r = 0; r < L_ / 256; ++r)
    w[(size_t)b * L_ + tid + r * 256] = vals[r] * inv;
}

// output[b,i,z] = w[b,i] * s[b,z]   (67 MB of float4 streaming stores)
__global__ void out_kernel(const float* __restrict__ w, const float* __restrict__ s,
                           float* __restrict__ out) {
  const size_t idx4 = (size_t)blockIdx.x * 256 + threadIdx.x;   // float4 index
  const int    z4   = (int)(idx4 & (Z_ / 4 - 1));               // Z/4 = 128
  const size_t bi   = idx4 >> 7;
  const int    b    = (int)(bi >> 11);
  const float  wv   = w[bi];
  const float4 sv   = ((const float4*)s)[(size_t)b * (Z_ / 4) + z4];
  float4 o;
  o.x = wv * sv.x; o.y = wv * sv.y; o.z = wv * sv.z; o.w = wv * sv.w;
  ((float4*)out)[idx4] = o;
}

// attn[b,i,j] = w[b,i]   (268 MB of splat float4 stores — the roofline limiter)
__global__ void attn_kernel(const float* __restrict__ w, float* __restrict__ attn) {
  const size_t idx4 = (size_t)blockIdx.x * 256 + threadIdx.x;   // float4 index
  const size_t row  = idx4 >> 9;                                // L/4 = 512
  const float  wv   = w[row];
  const float4 o = {wv, wv, wv, wv};
  ((float4*)attn)[idx4] = o;
}

extern "C" void kernel_launch(void* const* d_in, const int* in_sizes, int n_in,
                              void* d_out, int out_size, void* d_ws, size_t ws_size,
                              hipStream_t stream) {
  const float* enc  = (const float*)d_in[0];   // [B,L,H]
  const float* fin  = (const float*)d_in[1];   // [B,H]
  const float* zseq = (const float*)d_in[2];   // [B,L,Z]
  const int*   mask = (const int*)d_in[3];     // [B,L]
  const float* wq   = (const float*)d_in[4];   // [H,H]
  const float* wk   = (const float*)d_in[5];   // [H,H]
  const float* wv   = (const float*)d_in[6];   // [Z,Z]

  float* out  = (float*)d_out;                       // [B,L,Z]
  float* attn = out + (size_t)B_ * L_ * Z_;          // [B,L,L]

  float* ws   = (float*)d_ws;                        // ~459 KB used
  float* zsum = ws;                                  // [B,Z]   8192
  float* kmat = zsum + B_ * Z_;                      // [B,H]  16384
  float* kq   = kmat + B_ * H_;                      // [B,H]  16384
  float* svec = kq + B_ * H_;                        // [B,Z]   8192
  float* algn = svec + B_ * Z_;                      // [B,L]  32768
  float* wsm  = algn + B_ * L_;                      // [B,L]  32768
  float* part = attn;   // reuse attn output region as scratch; fully overwritten later

  // Σ_j z[b,j,:]  (two-stage, no atomics)
  zsum_partial<<<dim3(Z_ / 256, 16, B_), 256, 0, stream>>>(zseq, part);
  zsum_reduce<<<dim3(Z_ / 256, B_), 256, 0, stream>>>(part, zsum);

  // k = final @ W_k          [16,H] = [16,H][H,H]
  wmma_gemm16<0><<<H_ / 16, 32, 0, stream>>>(fin, H_, wk, H_, kmat, H_, H_);
  // kq = k @ W_q^T           [16,H] = [16,H][H,H]^T
  wmma_gemm16<1><<<H_ / 16, 32, 0, stream>>>(kmat, H_, wq, H_, kq, H_, H_);
  // s = zsum @ W_v           [16,Z] = [16,Z][Z,Z]
  wmma_gemm16<0><<<Z_ / 16, 32, 0, stream>>>(zsum, Z_, wv, Z_, svec, Z_, Z_);

  // align[b,i] = enc[b,i].kq[b] / 32
  align_kernel<<<(B_ * L_) / 8, 256, 0, stream>>>(enc, kq, algn);
  // w[b,:] = masked softmax over i
  softmax_kernel<<<B_, 256, 0, stream>>>(algn, mask, wsm);

  // outputs
  out_kernel<<<(B_ * L_ * Z_ / 4) / 256, 256, 0, stream>>>(wsm, svec, out);
  attn_kernel<<<(unsigned)((size_t)B_ * L_ * L_ / 4 / 256), 256, 0, stream>>>(wsm, attn);
}